// AdaMixerDecoderStage_77249281786080
// MI455X (gfx1250) — compile-verified
//
#include <hip/hip_runtime.h>
#include <math.h>

// ---------------------------------------------------------------------------
// AdaMixer decoder stage for MI455X (gfx1250), fp32 throughout.
// Dense math via V_WMMA_F32_16X16X4_F32 (wave32 WMMA), memory-bound workload.
// ---------------------------------------------------------------------------

typedef __attribute__((ext_vector_type(2))) float v2f;
typedef __attribute__((ext_vector_type(8))) float v8f;

#define B_    2
#define N_    500
#define C_    256
#define NH_   8
#define HD_   32
#define G_    4
#define PIN_  32
#define POUT_ 128
#define FFN_  2048
#define NCLS_ 80
#define CG_   64
#define BN_   (B_ * N_)         // 1000
#define PGN_  32768             // G*(CG*CG + PIN*POUT)

__device__ __forceinline__ v8f wmma_f32(v2f a, v2f b, v8f c) {
  // D = A(16x4,f32) * B(4x16,f32) + C(16x16,f32)
  return __builtin_amdgcn_wmma_f32_16x16x4_f32(
      /*neg_a=*/false, a, /*neg_b=*/false, b,
      /*c_mod=*/(short)0, c, /*reuse_a=*/false, /*reuse_b=*/false);
}

// ---------------------------------------------------------------------------
// Generic WMMA GEMM: Cout(MxN) = A(MxK) @ W(KxN) + bias, optional ReLU.
// Block = 256 threads = 8 waves -> 32x64 output tile (2x4 grid of 16x16).
// ---------------------------------------------------------------------------
__global__ __launch_bounds__(256) void gemm_wmma(
    const float* __restrict__ A, const float* __restrict__ W,
    const float* __restrict__ bias, float* __restrict__ Cout,
    int M, int N, int K, int act)
{
  __shared__ float As[32][33];
  __shared__ float Ws[32][68];

  const int tid   = threadIdx.x;
  const int lane  = tid & 31;
  const int wv    = tid >> 5;
  const int waveM = wv >> 2;         // 0..1
  const int waveN = wv & 3;          // 0..3
  const int hh    = lane >> 4;       // lane half (0/1)
  const int l16   = lane & 15;
  const int koff  = hh * 2;          // K sub-offset per ISA A/B layout
  const int tileM0 = blockIdx.y * 32;
  const int tileN0 = blockIdx.x * 64;

  v8f acc = {};
  for (int kb = 0; kb < K; kb += 32) {
    for (int idx = tid; idx < 32 * 32; idx += 256) {
      int r = idx >> 5, c = idx & 31;
      int m = tileM0 + r;
      As[r][c] = (m < M) ? A[m * K + kb + c] : 0.0f;
    }
    for (int idx = tid; idx < 32 * 64; idx += 256) {
      int r = idx >> 6, c = idx & 63;
      int nn = tileN0 + c;
      Ws[r][c] = (nn < N) ? W[(kb + r) * N + nn] : 0.0f;
    }
    if (kb + 32 < K)  // prefetch next weight slab (global_prefetch_b8)
      __builtin_prefetch(&W[(size_t)(kb + 32) * N + tileN0 + (tid & 63)], 0, 1);
    __syncthreads();
#pragma unroll
    for (int kk = 0; kk < 8; ++kk) {
      int k0 = kk * 4 + koff;
      v2f a; a.x = As[waveM * 16 + l16][k0];  a.y = As[waveM * 16 + l16][k0 + 1];
      v2f b; b.x = Ws[k0][waveN * 16 + l16];  b.y = Ws[k0 + 1][waveN * 16 + l16];
      acc = wmma_f32(a, b, acc);
    }
    __syncthreads();
  }
  int col = tileN0 + waveN * 16 + l16;
  if (col < N) {
    float bv = bias ? bias[col] : 0.0f;
#pragma unroll
    for (int i = 0; i < 8; ++i) {
      int row = tileM0 + waveM * 16 + hh * 8 + i;   // C/D layout: VGPR i -> M = i (+8 for hi half)
      if (row < M) {
        float v = acc[i] + bv;
        if (act == 1) v = fmaxf(v, 0.0f);
        Cout[(size_t)row * N + col] = v;
      }
    }
  }
}

// ---------------------------------------------------------------------------
// Positional embedding + qa = content + pe, and ROI decode.
// ---------------------------------------------------------------------------
__global__ __launch_bounds__(256) void prep_kernel(
    const float* __restrict__ xyzr, const float* __restrict__ qcont,
    float* __restrict__ qa, float* __restrict__ rois)
{
  int row = blockIdx.x;  // b*N + n
  int c = threadIdx.x;
  float x = xyzr[row * 4 + 0], y = xyzr[row * 4 + 1];
  float z = xyzr[row * 4 + 2], r = xyzr[row * 4 + 3];
  float tok[4] = { x * 0.001f, y * 0.001f, z, r };
  int comp = c >> 6;
  int k = (c & 63) >> 1;
  float t = __powf(10000.0f, (float)k / 32.0f);
  float val = tok[comp] / t;
  float pe = (c & 1) ? __cosf(val) : __sinf(val);
  qa[row * C_ + c] = qcont[row * C_ + c] + pe;
  if (c == 0) {
    float sc = exp2f(z);
    float rw = sc * exp2f(-0.5f * r);
    float rh = sc * exp2f(0.5f * r);
    rois[row * 4 + 0] = x - 0.5f * rw;
    rois[row * 4 + 1] = y - 0.5f * rh;
    rois[row * 4 + 2] = x + 0.5f * rw;
    rois[row * 4 + 3] = y + 0.5f * rh;
  }
}

// log(IoF + 1e-7), one block per (b, i) row.
__global__ __launch_bounds__(256) void iof_kernel(
    const float* __restrict__ rois, float* __restrict__ iofLog)
{
  int b = blockIdx.x / N_;
  int i = blockIdx.x % N_;
  const float* bi = rois + (size_t)(b * N_ + i) * 4;
  float x1 = bi[0], y1 = bi[1], x2 = bi[2], y2 = bi[3];
  float area1 = (x2 - x1) * (y2 - y1);
  for (int j = threadIdx.x; j < N_; j += blockDim.x) {
    const float* bj = rois + (size_t)(b * N_ + j) * 4;
    float lx = fmaxf(x1, bj[0]), ly = fmaxf(y1, bj[1]);
    float rx = fminf(x2, bj[2]), ry = fminf(y2, bj[3]);
    float ow = fmaxf(rx - lx, 0.0f), oh = fmaxf(ry - ly, 0.0f);
    float iof = (ow * oh) / fmaxf(area1, 1e-6f);
    iofLog[((size_t)(b * N_ + i)) * N_ + j] = __logf(iof + 1e-7f);
  }
}

// logits[b,h,i,j] = scale * q.k + tau[h] * iofLog[b,i,j]
__global__ __launch_bounds__(128) void attn_logits_kernel(
    const float* __restrict__ qkv, const float* __restrict__ iofLog,
    const float* __restrict__ tau, float* __restrict__ logits)
{
  int bh = blockIdx.x / N_;
  int i  = blockIdx.x % N_;
  int b = bh / NH_, h = bh % NH_;
  __shared__ float qrow[HD_];
  if (threadIdx.x < HD_)
    qrow[threadIdx.x] =
        qkv[(size_t)(b * N_ + i) * (3 * C_) + h * HD_ + threadIdx.x] * 0.17677669529663687f;
  __syncthreads();
  float th = tau[h];
  for (int j = threadIdx.x; j < N_; j += blockDim.x) {
    const float* kv = qkv + (size_t)(b * N_ + j) * (3 * C_) + C_ + h * HD_;
    float d = 0.0f;
#pragma unroll
    for (int dd = 0; dd < HD_; ++dd) d += qrow[dd] * kv[dd];
    logits[(size_t)blockIdx.x * N_ + j] = d + th * iofLog[((size_t)(b * N_ + i)) * N_ + j];
  }
}

__global__ __launch_bounds__(256) void softmax_kernel(float* __restrict__ data, int n)
{
  __shared__ float red[256];
  float* rowp = data + (size_t)blockIdx.x * n;
  int tid = threadIdx.x;
  float m = -1e30f;
  for (int j = tid; j < n; j += 256) m = fmaxf(m, rowp[j]);
  red[tid] = m; __syncthreads();
  for (int s = 128; s > 0; s >>= 1) { if (tid < s) red[tid] = fmaxf(red[tid], red[tid + s]); __syncthreads(); }
  m = red[0]; __syncthreads();
  float sum = 0.0f;
  for (int j = tid; j < n; j += 256) { float e = __expf(rowp[j] - m); rowp[j] = e; sum += e; }
  red[tid] = sum; __syncthreads();
  for (int s = 128; s > 0; s >>= 1) { if (tid < s) red[tid] += red[tid + s]; __syncthreads(); }
  float inv = 1.0f / red[0];
  for (int j = tid; j < n; j += 256) rowp[j] *= inv;
}

// o[b,i, h*32+d] = sum_j attn[b,h,i,j] * v[b,j,h*32+d]
__global__ __launch_bounds__(256) void attn_av_kernel(
    const float* __restrict__ attn, const float* __restrict__ qkv, float* __restrict__ o)
{
  int row = blockIdx.x;          // b*N + i
  int b = row / N_;
  int h = threadIdx.x >> 5;
  int d = threadIdx.x & 31;
  const float* ar = attn + (((size_t)(b * NH_ + h)) * N_ + (row % N_)) * N_;
  float acc = 0.0f;
  for (int j = 0; j < N_; ++j)
    acc += ar[j] * qkv[(size_t)(b * N_ + j) * (3 * C_) + 2 * C_ + h * HD_ + d];
  o[(size_t)row * C_ + h * HD_ + d] = acc;
}

// LayerNorm over C=256 with optional residual / affine / relu.
__global__ __launch_bounds__(256) void ln_kernel(
    const float* __restrict__ x, const float* __restrict__ res,
    const float* __restrict__ g, const float* __restrict__ bta,
    float* __restrict__ out, int relu)
{
  __shared__ float red[256];
  int row = blockIdx.x, c = threadIdx.x;
  float v = x[(size_t)row * C_ + c];
  if (res) v += res[(size_t)row * C_ + c];
  red[c] = v; __syncthreads();
  for (int s = 128; s > 0; s >>= 1) { if (c < s) red[c] += red[c + s]; __syncthreads(); }
  float mu = red[0] * (1.0f / C_); __syncthreads();
  float dv = v - mu;
  red[c] = dv * dv; __syncthreads();
  for (int s = 128; s > 0; s >>= 1) { if (c < s) red[c] += red[c + s]; __syncthreads(); }
  float var = red[0] * (1.0f / C_);
  float y = dv * rsqrtf(var + 1e-5f);
  if (g) y = y * g[c] + bta[c];
  if (relu) y = fmaxf(y, 0.0f);
  out[(size_t)row * C_ + c] = y;
}

// Sampling point coords + level softmax weights.
__global__ __launch_bounds__(128) void samp_prep_kernel(
    const float* __restrict__ xyzr, const float* __restrict__ off,
    float* __restrict__ sxy, float* __restrict__ lw)
{
  int row = blockIdx.x;
  int p = threadIdx.x;  // 0..127
  float qx = xyzr[row * 4 + 0], qy = xyzr[row * 4 + 1];
  float qz = xyzr[row * 4 + 2], qr = xyzr[row * 4 + 3];
  float sc = exp2f(qz);
  float rw = sc * exp2f(-0.5f * qr);
  float rh = sc * exp2f(0.5f * qr);
  const float* op = off + (size_t)row * (G_ * PIN_ * 3) + p * 3;
  float sx = qx + op[0] * rw;
  float sy = qy + op[1] * rh;
  float lz = qz + op[2] - 3.0f;
  float e[4], m = -1e30f;
#pragma unroll
  for (int l = 0; l < 4; ++l) { float dl = lz - (float)l; e[l] = -0.5f * dl * dl; m = fmaxf(m, e[l]); }
  float s = 0.0f;
#pragma unroll
  for (int l = 0; l < 4; ++l) { e[l] = __expf(e[l] - m); s += e[l]; }
  float inv = 1.0f / s;
  sxy[(size_t)row * 256 + p * 2 + 0] = sx;
  sxy[(size_t)row * 256 + p * 2 + 1] = sy;
#pragma unroll
  for (int l = 0; l < 4; ++l) lw[(size_t)row * 512 + p * 4 + l] = e[l] * inv;
}

// Multi-level bilinear gather -> feat (B,N,G,PIN,CG).
__global__ __launch_bounds__(256) void sample_kernel(
    const float* __restrict__ x0, const float* __restrict__ x1,
    const float* __restrict__ x2, const float* __restrict__ x3,
    const float* __restrict__ sxy, const float* __restrict__ lw,
    float* __restrict__ feat)
{
  const int   Hs[4]   = {192, 96, 48, 24};
  const int   Wsz[4]  = {336, 168, 84, 42};
  const float Sinv[4] = {0.25f, 0.125f, 0.0625f, 0.03125f};
  const float* xs[4] = {x0, x1, x2, x3};

  int bng = blockIdx.x;
  int g = bng % G_;
  int row = bng / G_;           // b*N + n
  int b = row / N_;
  int d = threadIdx.x & 63;
  int psub = threadIdx.x >> 6;
  int ch = g * CG_ + d;
  for (int p = psub; p < PIN_; p += 4) {
    int pidx = g * PIN_ + p;
    float sx = sxy[(size_t)row * 256 + pidx * 2 + 0];
    float sy = sxy[(size_t)row * 256 + pidx * 2 + 1];
    float acc = 0.0f;
#pragma unroll
    for (int l = 0; l < 4; ++l) {
      float wl = lw[(size_t)row * 512 + pidx * 4 + l];
      int H = Hs[l], W = Wsz[l];
      float px = sx * Sinv[l] - 0.5f;
      float py = sy * Sinv[l] - 0.5f;
      float fx = floorf(px), fy = floorf(py);
      float wx = px - fx, wy = py - fy;
      int xi = (int)fx, yi = (int)fy;
      const float* base = xs[l] + ((size_t)(b * C_ + ch)) * H * W;
      float v00 = (xi >= 0 && xi < W     && yi >= 0 && yi < H)     ? base[yi * W + xi]           : 0.0f;
      float v10 = (xi + 1 >= 0 && xi + 1 < W && yi >= 0 && yi < H) ? base[yi * W + xi + 1]       : 0.0f;
      float v01 = (xi >= 0 && xi < W && yi + 1 >= 0 && yi + 1 < H) ? base[(yi + 1) * W + xi]     : 0.0f;
      float v11 = (xi + 1 >= 0 && xi + 1 < W && yi + 1 >= 0 && yi + 1 < H) ? base[(yi + 1) * W + xi + 1] : 0.0f;
      acc += wl * (v00 * (1 - wx) * (1 - wy) + v10 * wx * (1 - wy) +
                   v01 * (1 - wx) * wy       + v11 * wx * wy);
    }
    feat[(size_t)bng * 2048 + p * 64 + d] = acc;
  }
}

// Mixing stage 1: out1 = feat(32x64) @ Mm(64x64); fused ln2d + relu.
__global__ __launch_bounds__(256) void mix1_kernel(
    const float* __restrict__ feat, const float* __restrict__ pg,
    float* __restrict__ mixed1)
{
  __shared__ float Fs[32][68];
  __shared__ float Ms[64][68];
  __shared__ float Os[32][68];
  __shared__ float red[256];
  __shared__ float red2[256];
  int bng = blockIdx.x;
  int tid = threadIdx.x, lane = tid & 31, wv = tid >> 5;
  int waveM = wv >> 2, waveN = wv & 3;
  int hh = lane >> 4, l16 = lane & 15, koff = hh * 2;

  for (int idx = tid; idx < 2048; idx += 256)
    Fs[idx >> 6][idx & 63] = feat[(size_t)bng * 2048 + idx];
  for (int idx = tid; idx < 4096; idx += 256)
    Ms[idx >> 6][idx & 63] = pg[(size_t)bng * 8192 + idx];
  __syncthreads();

  v8f acc = {};
#pragma unroll
  for (int kk = 0; kk < 16; ++kk) {
    int k0 = kk * 4 + koff;
    v2f a; a.x = Fs[waveM * 16 + l16][k0]; a.y = Fs[waveM * 16 + l16][k0 + 1];
    v2f b; b.x = Ms[k0][waveN * 16 + l16]; b.y = Ms[k0 + 1][waveN * 16 + l16];
    acc = wmma_f32(a, b, acc);
  }
  int col = waveN * 16 + l16;
#pragma unroll
  for (int i = 0; i < 8; ++i)
    Os[waveM * 16 + hh * 8 + i][col] = acc[i];
  __syncthreads();

  float s = 0.0f, s2 = 0.0f;
  for (int e = tid; e < 2048; e += 256) { float v = Os[e >> 6][e & 63]; s += v; s2 += v * v; }
  red[tid] = s; red2[tid] = s2; __syncthreads();
  for (int st = 128; st > 0; st >>= 1) {
    if (tid < st) { red[tid] += red[tid + st]; red2[tid] += red2[tid + st]; }
    __syncthreads();
  }
  float mu = red[0] / 2048.0f;
  float var = red2[0] / 2048.0f - mu * mu;
  float rs = rsqrtf(var + 1e-5f);
  for (int e = tid; e < 2048; e += 256) {
    float v = (Os[e >> 6][e & 63] - mu) * rs;
    mixed1[(size_t)bng * 2048 + e] = fmaxf(v, 0.0f);
  }
}

// Mixing stage 2: out2 = Ss(128x32) @ mixed1(32x64); fused ln2d + relu.
__global__ __launch_bounds__(256) void mix2_kernel(
    const float* __restrict__ pg, const float* __restrict__ mixed1,
    float* __restrict__ out2)
{
  __shared__ float Ssh[128][36];
  __shared__ float Xs[32][68];
  __shared__ float Os[128][68];
  __shared__ float red[256];
  __shared__ float red2[256];
  int bng = blockIdx.x;
  int g = bng % G_;
  int row = bng / G_;
  int tid = threadIdx.x, lane = tid & 31, wv = tid >> 5;
  int hh = lane >> 4, l16 = lane & 15, koff = hh * 2;

  for (int idx = tid; idx < 4096; idx += 256)
    Ssh[idx >> 5][idx & 31] = pg[(size_t)bng * 8192 + 4096 + idx];
  for (int idx = tid; idx < 2048; idx += 256)
    Xs[idx >> 6][idx & 63] = mixed1[(size_t)bng * 2048 + idx];
  __syncthreads();

#pragma unroll
  for (int it = 0; it < 4; ++it) {
    int tt = wv + 8 * it;
    int tileM = tt >> 2, tileN = tt & 3;
    v8f acc = {};
#pragma unroll
    for (int kk = 0; kk < 8; ++kk) {
      int k0 = kk * 4 + koff;
      v2f a; a.x = Ssh[tileM * 16 + l16][k0]; a.y = Ssh[tileM * 16 + l16][k0 + 1];
      v2f b; b.x = Xs[k0][tileN * 16 + l16]; b.y = Xs[k0 + 1][tileN * 16 + l16];
      acc = wmma_f32(a, b, acc);
    }
    int col = tileN * 16 + l16;
#pragma unroll
    for (int i = 0; i < 8; ++i)
      Os[tileM * 16 + hh * 8 + i][col] = acc[i];
  }
  __syncthreads();

  float s = 0.0f, s2 = 0.0f;
  for (int e = tid; e < 8192; e += 256) { float v = Os[e >> 6][e & 63]; s += v; s2 += v * v; }
  red[tid] = s; red2[tid] = s2; __syncthreads();
  for (int st = 128; st > 0; st >>= 1) {
    if (tid < st) { red[tid] += red[tid + st]; red2[tid] += red2[tid + st]; }
    __syncthreads();
  }
  float mu = red[0] / 8192.0f;
  float var = red2[0] / 8192.0f - mu * mu;
  float rs = rsqrtf(var + 1e-5f);
  float* dst = out2 + (size_t)row * PGN_ + (size_t)g * 8192;
  for (int e = tid; e < 8192; e += 256) {
    float v = (Os[e >> 6][e & 63] - mu) * rs;
    dst[e] = fmaxf(v, 0.0f);
  }
}

// ---------------------------------------------------------------------------
extern "C" void kernel_launch(void* const* d_in, const int* in_sizes, int n_in,
                              void* d_out, int out_size, void* d_ws, size_t ws_size,
                              hipStream_t stream)
{
  (void)in_sizes; (void)n_in; (void)out_size; (void)ws_size;
  const float* x0    = (const float*)d_in[0];
  const float* x1    = (const float*)d_in[1];
  const float* x2    = (const float*)d_in[2];
  const float* x3    = (const float*)d_in[3];
  const float* xyzr  = (const float*)d_in[4];
  const float* qcont = (const float*)d_in[5];
  const float* tau       = (const float*)d_in[6];
  const float* in_proj_w = (const float*)d_in[7];
  const float* in_proj_b = (const float*)d_in[8];
  const float* attn_out_w = (const float*)d_in[9];
  const float* attn_out_b = (const float*)d_in[10];
  const float* attn_ln_g  = (const float*)d_in[11];
  const float* attn_ln_b  = (const float*)d_in[12];
  const float* off_w = (const float*)d_in[13];
  const float* off_b = (const float*)d_in[14];
  const float* pg_w  = (const float*)d_in[15];
  const float* pg_b  = (const float*)d_in[16];
  const float* mix_out_w = (const float*)d_in[17];
  const float* mix_out_b = (const float*)d_in[18];
  const float* asm_ln_g = (const float*)d_in[19];
  const float* asm_ln_b = (const float*)d_in[20];
  const float* ffn_w1 = (const float*)d_in[21];
  const float* ffn_b1 = (const float*)d_in[22];
  const float* ffn_w2 = (const float*)d_in[23];
  const float* ffn_b2 = (const float*)d_in[24];
  const float* ffn_ln_g = (const float*)d_in[25];
  const float* ffn_ln_b = (const float*)d_in[26];
  const float* cls_w    = (const float*)d_in[27];
  const float* cls_ln_g = (const float*)d_in[28];
  const float* cls_ln_b = (const float*)d_in[29];
  const float* fc_cls_w = (const float*)d_in[30];
  const float* fc_cls_b = (const float*)d_in[31];
  const float* reg_w    = (const float*)d_in[32];
  const float* reg_ln_g = (const float*)d_in[33];
  const float* reg_ln_b = (const float*)d_in[34];
  const float* fc_reg_w = (const float*)d_in[35];
  const float* fc_reg_b = (const float*)d_in[36];

  float* ws = (float*)d_ws;
  size_t off = 0;
  auto alloc = [&](size_t n) { float* p = ws + off; off += n; return p; };
  float* qa      = alloc((size_t)BN_ * C_);
  float* rois    = alloc((size_t)BN_ * 4);
  float* iofLog  = alloc((size_t)B_ * N_ * N_);
  float* qkv     = alloc((size_t)BN_ * 3 * C_);
  float* logits  = alloc((size_t)B_ * NH_ * N_ * N_);
  float* oatt    = alloc((size_t)BN_ * C_);
  float* attnpj  = alloc((size_t)BN_ * C_);
  float* qc1     = alloc((size_t)BN_ * C_);
  float* offbuf  = alloc((size_t)BN_ * G_ * PIN_ * 3);
  float* sxy     = alloc((size_t)BN_ * 256);
  float* lw      = alloc((size_t)BN_ * 512);
  float* feat    = alloc((size_t)BN_ * G_ * PIN_ * CG_);
  float* pg      = alloc((size_t)BN_ * PGN_);
  float* mixed1  = alloc((size_t)BN_ * G_ * PIN_ * CG_);
  float* out2    = alloc((size_t)BN_ * PGN_);
  float* mixtmp  = alloc((size_t)BN_ * C_);
  float* qc2     = alloc((size_t)BN_ * C_);
  float* ffnh    = alloc((size_t)BN_ * FFN_);
  float* ffntmp  = alloc((size_t)BN_ * C_);
  float* qc3     = alloc((size_t)BN_ * C_);
  float* t0      = alloc((size_t)BN_ * C_);
  float* t1      = alloc((size_t)BN_ * C_);

  auto gemm = [&](const float* A, const float* W, const float* bias, float* Cc,
                  int M, int N, int K, int act) {
    dim3 grid((N + 63) / 64, (M + 31) / 32);
    gemm_wmma<<<grid, 256, 0, stream>>>(A, W, bias, Cc, M, N, K, act);
  };

  // Attention prologue
  prep_kernel<<<BN_, 256, 0, stream>>>(xyzr, qcont, qa, rois);
  iof_kernel<<<BN_, 256, 0, stream>>>(rois, iofLog);
  gemm(qa, in_proj_w, in_proj_b, qkv, BN_, 3 * C_, C_, 0);
  attn_logits_kernel<<<B_ * NH_ * N_, 128, 0, stream>>>(qkv, iofLog, tau, logits);
  softmax_kernel<<<B_ * NH_ * N_, 256, 0, stream>>>(logits, N_);
  attn_av_kernel<<<BN_, 256, 0, stream>>>(logits, qkv, oatt);
  gemm(oatt, attn_out_w, attn_out_b, attnpj, BN_, C_, C_, 0);
  ln_kernel<<<BN_, 256, 0, stream>>>(attnpj, qa, attn_ln_g, attn_ln_b, qc1, 0);

  // Adaptive sampling
  gemm(qc1, off_w, off_b, offbuf, BN_, G_ * PIN_ * 3, C_, 0);
  samp_prep_kernel<<<BN_, 128, 0, stream>>>(xyzr, offbuf, sxy, lw);
  sample_kernel<<<BN_ * G_, 256, 0, stream>>>(x0, x1, x2, x3, sxy, lw, feat);

  // Parameter generator + adaptive mixing
  gemm(qc1, pg_w, pg_b, pg, BN_, PGN_, C_, 0);
  mix1_kernel<<<BN_ * G_, 256, 0, stream>>>(feat, pg, mixed1);
  mix2_kernel<<<BN_ * G_, 256, 0, stream>>>(pg, mixed1, out2);
  gemm(out2, mix_out_w, mix_out_b, mixtmp, BN_, C_, PGN_, 0);
  ln_kernel<<<BN_, 256, 0, stream>>>(mixtmp, qc1, asm_ln_g, asm_ln_b, qc2, 0);

  // FFN
  gemm(qc2, ffn_w1, ffn_b1, ffnh, BN_, FFN_, C_, 1);
  gemm(ffnh, ffn_w2, ffn_b2, ffntmp, BN_, C_, FFN_, 0);
  ln_kernel<<<BN_, 256, 0, stream>>>(ffntmp, qc2, ffn_ln_g, ffn_ln_b, qc3, 0);

  // Heads
  float* out_cls = (float*)d_out;                 // (B,N,80)
  float* out_reg = (float*)d_out + BN_ * NCLS_;   // (B,N,4)
  float* out_qc  = out_reg + BN_ * 4;             // (B,N,C)

  gemm(qc3, cls_w, nullptr, t0, BN_, C_, C_, 0);
  ln_kernel<<<BN_, 256, 0, stream>>>(t0, nullptr, cls_ln_g, cls_ln_b, t1, 1);
  gemm(t1, fc_cls_w, fc_cls_b, out_cls, BN_, NCLS_, C_, 0);

  const float* regin = qc3;
  for (int i = 0; i < 3; ++i) {
    gemm(regin, reg_w + (size_t)i * C_ * C_, nullptr, t0, BN_, C_, C_, 0);
    ln_kernel<<<BN_, 256, 0, stream>>>(t0, nullptr, reg_ln_g + i * C_, reg_ln_b + i * C_, t1, 1);
    regin = t1;
  }
  gemm(regin, fc_reg_w, fc_reg_b, out_reg, BN_, 4, C_, 0);

  hipMemcpyAsync(out_qc, qc3, (size_t)BN_ * C_ * sizeof(float),
                 hipMemcpyDeviceToDevice, stream);
}